// HebbFF_22789096472916
// MI455X (gfx1250) — compile-verified
//
#include <hip/hip_runtime.h>

typedef __attribute__((ext_vector_type(2))) float v2f;
typedef __attribute__((ext_vector_type(8))) float v8f;

#define T_ 16
#define B_ 128
#define N_ 512
#define D_ 256

// ---------------------------------------------------------------------------
// Kernel A: P[(t*B+b), n] = sum_d w1[n,d] * x[(t*B+b), d]
// GEMM M=512 (n), N'=2048 (t*B+b), K=256 via V_WMMA_F32_16X16X4_F32.
// One wave owns one 16x16 output tile; 32x128 = 4096 tiles = 512 blocks x 8 waves.
// f32 WMMA fragment layout (ISA 7.12.2): lane L<16 holds (K=k,k+1), lane L>=16
// holds (K=k+2,k+3) -> one float2 load per lane per K-step.
// ---------------------------------------------------------------------------
__global__ __launch_bounds__(256) void hebb_w1x_gemm(const float* __restrict__ w1,
                                                     const float* __restrict__ x,
                                                     float* __restrict__ P) {
    const int wid  = blockIdx.x * (blockDim.x >> 5) + (threadIdx.x >> 5);
    const int lane = threadIdx.x & 31;
    const int lr   = lane & 15;
    const int hi   = lane >> 4;

    const int tile_n = wid & 31;   // 32 tiles across N=512
    const int tile_c = wid >> 5;   // 128 tiles across T*B=2048
    const int n0 = tile_n << 4;
    const int c0 = tile_c << 4;

    const float* __restrict__ wrow = w1 + (size_t)(n0 + lr) * D_;
    const float* __restrict__ xrow = x  + (size_t)(c0 + lr) * D_;

    v8f acc = {};
#pragma unroll 4
    for (int k = 0; k < D_; k += 4) {
        const int kk = k + 2 * hi;
        v2f a = *(const v2f*)(wrow + kk);
        v2f b = *(const v2f*)(xrow + kk);
        acc = __builtin_amdgcn_wmma_f32_16x16x4_f32(
            /*neg_a=*/false, a, /*neg_b=*/false, b,
            /*c_mod=*/(short)0, acc, /*reuse_a=*/false, /*reuse_b=*/false);
    }

    // C/D layout: lane L holds column N=L&15, VGPR r holds row M = r + 8*(L>>4)
    float* __restrict__ orow = P + (size_t)(c0 + lr) * N_ + n0 + 8 * hi;
#pragma unroll
    for (int r = 0; r < 8; ++r) orow[r] = acc[r];
}

// ---------------------------------------------------------------------------
// Kernel B: G[b, t, s] = sum_d x[t,b,d] * x[s,b,d]   (16x16 per batch)
// One wave per batch; A-fragment == B-fragment (both index time with same K
// layout), so a single float2 load feeds both WMMA operands.
// ---------------------------------------------------------------------------
__global__ __launch_bounds__(128) void hebb_gram(const float* __restrict__ x,
                                                 float* __restrict__ G) {
    const int b    = blockIdx.x * (blockDim.x >> 5) + (threadIdx.x >> 5); // 0..127
    const int lane = threadIdx.x & 31;
    const int lr   = lane & 15;
    const int hi   = lane >> 4;

    const float* __restrict__ base = x + (size_t)(lr * B_ + b) * D_;

    v8f acc = {};
#pragma unroll 4
    for (int k = 0; k < D_; k += 4) {
        v2f f = *(const v2f*)(base + k + 2 * hi);
        acc = __builtin_amdgcn_wmma_f32_16x16x4_f32(
            false, f, false, f, (short)0, acc, false, false);
    }

    float* __restrict__ g = G + (size_t)b * (T_ * T_);
#pragma unroll
    for (int r = 0; r < 8; ++r) g[(r + 8 * hi) * T_ + lr] = acc[r];
}

// ---------------------------------------------------------------------------
// Kernel C: sequential scan over T. One block per batch, one thread per n.
//   pre[n]  = P[t,b,n] + b1 + eta * sum_{s<t} lam^{t-1-s} G[b,s,t] h_s[n]
//   h_t     = sigmoid(pre);  H kept in LDS (16x512 fp32 = 32KB)
//   y[t,b]  = sigmoid(w2 * sum_n h_t[n] + b2)   (wave32 shuffle + LDS reduce)
// ---------------------------------------------------------------------------
__global__ __launch_bounds__(512) void hebb_scan(const float* __restrict__ P,
                                                 const float* __restrict__ G,
                                                 const float* __restrict__ b1_param,
                                                 const float* __restrict__ b2,
                                                 const float* __restrict__ lmbda,
                                                 const float* __restrict__ eta,
                                                 const float* __restrict__ w2_param,
                                                 float* __restrict__ out) {
    __shared__ float Hs[T_ * N_];     // 32 KB
    __shared__ float Gs[T_ * T_];     // 1 KB
    __shared__ float partial[16];

    const int b   = blockIdx.x;
    const int tid = threadIdx.x;

    const float b1v = b1_param[0];
    const float b2v = b2[0];
    const float lam = 1.0f / (1.0f + __expf(-lmbda[0]));
    const float et  = eta[0];
    const float w2v = w2_param[0];

    if (tid < T_ * T_) Gs[tid] = G[(size_t)b * (T_ * T_) + tid];
    __syncthreads();

    for (int t = 0; t < T_; ++t) {
        // recurrent term: lam^{t-1-s} built incrementally, newest-first
        float acc  = 0.0f;
        float lamp = 1.0f;
        for (int s = t - 1; s >= 0; --s) {
            acc  += lamp * Gs[s * T_ + t] * Hs[s * N_ + tid];
            lamp *= lam;
        }

        const float pre = P[(size_t)(t * B_ + b) * N_ + tid] + b1v + et * acc;
        const float h   = 1.0f / (1.0f + __expf(-pre));
        Hs[t * N_ + tid] = h;

        // reduce sum_n h over 512 threads: wave32 shuffle then 16 partials
        float v = h;
#pragma unroll
        for (int off = 16; off > 0; off >>= 1) v += __shfl_down(v, off, 32);
        if ((tid & 31) == 0) partial[tid >> 5] = v;
        __syncthreads();
        if (tid == 0) {
            float s = 0.0f;
#pragma unroll
            for (int i = 0; i < 16; ++i) s += partial[i];
            out[t * B_ + b] = 1.0f / (1.0f + __expf(-(w2v * s + b2v)));
        }
        __syncthreads();  // Hs[t] visible + partial[] safe before next step
    }
}

// ---------------------------------------------------------------------------
extern "C" void kernel_launch(void* const* d_in, const int* in_sizes, int n_in,
                              void* d_out, int out_size, void* d_ws, size_t ws_size,
                              hipStream_t stream) {
    const float* x   = (const float*)d_in[0];   // [T,B,D]
    const float* w1  = (const float*)d_in[1];   // [N,D]
    const float* b1p = (const float*)d_in[2];   // scalar
    const float* b2  = (const float*)d_in[3];   // [1,1]
    const float* lmb = (const float*)d_in[4];   // scalar
    const float* eta = (const float*)d_in[5];   // scalar
    const float* w2p = (const float*)d_in[6];   // scalar
    float* out = (float*)d_out;                 // [T,B,1]

    float* P = (float*)d_ws;                    // [T*B, N]  = 4 MB
    float* G = P + (size_t)T_ * B_ * N_;        // [B, T, T] = 128 KB

    hebb_w1x_gemm<<<512, 256, 0, stream>>>(w1, x, P);
    hebb_gram<<<32, 128, 0, stream>>>(x, G);
    hebb_scan<<<B_, N_, 0, stream>>>(P, G, b1p, b2, lmb, eta, w2p, out);
}